// ZoeppritzTensor_16638703304896
// MI455X (gfx1250) — compile-verified
//
#include <hip/hip_runtime.h>

typedef __attribute__((ext_vector_type(2))) float v2f;
typedef __attribute__((ext_vector_type(8))) float v8f;

#define L_LAYERS 250
#define N_TRACES 600
#define N_THETA  30
#define KPAD     256   // layers padded for K/M tiling
#define APAD     32    // angles padded for N tiling
#define LDSW     260   // LDS row stride (floats): bank-conflict-free b64 loads

// ---------------------------------------------------------------------------
// Kernel 0: zero-pad wavemat (250x250) into Wpad (256x256) so GEMM A-loads
// need no bounds checks.
// ---------------------------------------------------------------------------
__global__ void zp_pad_w_kernel(const float* __restrict__ W, float* __restrict__ Wpad) {
    int idx = blockIdx.x * blockDim.x + threadIdx.x;
    if (idx >= KPAD * KPAD) return;
    int r = idx >> 8;
    int c = idx & 255;
    float v = 0.0f;
    if (r < L_LAYERS && c < L_LAYERS) v = W[r * L_LAYERS + c];
    Wpad[idx] = v;
}

// ---------------------------------------------------------------------------
// Kernel 1: Zoeppritz Rpp per (trace, interface, angle) written TRANSPOSED:
// RpadT[t][a][k], a in [0,32) (a>=30 zero), k in [0,256) (k>=249 zero).
// arcsin eliminated analytically: sin(asin(x))=x, cos(asin(x))=sqrt(1-x^2).
// 4x4 solve x0 via Cramer's rule (two determinants, shared minors).
// ---------------------------------------------------------------------------
__global__ void zp_rpp_kernel(const float* __restrict__ vp, const float* __restrict__ vs,
                              const float* __restrict__ rho, const float* __restrict__ theta,
                              float* __restrict__ RpadT) {
    int idx = blockIdx.x * blockDim.x + threadIdx.x;
    const int total = N_TRACES * KPAD * APAD;
    if (idx >= total) return;
    int a = idx & (APAD - 1);
    int k = (idx >> 5) & (KPAD - 1);
    int t = idx >> 13;   // 5 bits angle + 8 bits layer

    float val = 0.0f;
    if (k < (L_LAYERS - 1) && a < N_THETA) {
        float a1 = vp[k * N_TRACES + t],       a2 = vp[(k + 1) * N_TRACES + t];
        float b1 = vs[k * N_TRACES + t],       b2 = vs[(k + 1) * N_TRACES + t];
        float r1 = rho[k * N_TRACES + t],      r2 = rho[(k + 1) * N_TRACES + t];
        float th  = theta[a];
        float sth = __sinf(th);
        float cth = __cosf(th);
        float p   = sth / a1;

        float sth2 = fminf(1.0f, fmaxf(-1.0f, p * a2));
        float sp1  = fminf(1.0f, fmaxf(-1.0f, p * b1));
        float sp2  = fminf(1.0f, fmaxf(-1.0f, p * b2));
        float cth2 = __builtin_sqrtf(fmaxf(0.0f, 1.0f - sth2 * sth2));
        float cp1  = __builtin_sqrtf(fmaxf(0.0f, 1.0f - sp1 * sp1));
        float cp2  = __builtin_sqrtf(fmaxf(0.0f, 1.0f - sp2 * sp2));

        float w1 = 1.0f - 2.0f * sp1 * sp1;
        float w2 = 1.0f - 2.0f * sp2 * sp2;

        float m00 = -sth,                   m01 = -cp1,                   m02 = sth2,                   m03 = cp2;
        float m10 =  cth,                   m11 = -sp1,                   m12 = cth2,                   m13 = -sp2;
        float m20 =  2.0f*r1*b1*sp1*cth,    m21 = r1*b1*w1,               m22 = 2.0f*r2*b2*sp2*cth2,    m23 = r2*b2*w2;
        float m30 = -r1*a1*w1,              m31 = 2.0f*r1*b1*sp1*cp1,     m32 = r2*a2*w2,               m33 = -2.0f*r2*b2*sp2*cp2;
        float n0  =  sth, n1 = cth, n2 = m20, n3 = -m30;

        // shared 2x2 minors (cols 1..3 untouched by column-0 substitution)
        float s3 = m01*m12 - m02*m11;
        float s4 = m01*m13 - m03*m11;
        float s5 = m02*m13 - m03*m12;
        float c5 = m22*m33 - m23*m32;
        float c4 = m21*m33 - m23*m31;
        float c3 = m21*m32 - m22*m31;
        // det(M)
        float s0 = m00*m11 - m01*m10;
        float s1 = m00*m12 - m02*m10;
        float s2 = m00*m13 - m03*m10;
        float c2 = m20*m33 - m23*m30;
        float c1 = m20*m32 - m22*m30;
        float c0 = m20*m31 - m21*m30;
        float det  = s0*c5 - s1*c4 + s2*c3 + s3*c2 - s4*c1 + s5*c0;
        // det(M with col0 := n)
        float s0b = n0*m11 - m01*n1;
        float s1b = n0*m12 - m02*n1;
        float s2b = n0*m13 - m03*n1;
        float c2b = n2*m33 - m23*n3;
        float c1b = n2*m32 - m22*n3;
        float c0b = n2*m31 - m21*n3;
        float detX = s0b*c5 - s1b*c4 + s2b*c3 + s3*c2b - s4*c1b + s5*c0b;

        val = detX / det;
    }
    // transposed store: [t][a][k]
    RpadT[((size_t)t * APAD + a) * KPAD + k] = val;
}

// ---------------------------------------------------------------------------
// Kernel 2: per-trace GEMM  C(250x30) = W(250x250) x R(250x30)  in f32 via
// V_WMMA_F32_16X16X4_F32. 1 block per trace, 8 waves; each wave owns 2 M-tiles
// and both N-tiles, reusing the A fragment across N accumulators.
// R is staged transposed in LDS (row stride 260 floats) so each B fragment is
// one aligned 8-byte ds_load hitting all 64 banks exactly once per wave.
// ---------------------------------------------------------------------------
__global__ void __launch_bounds__(256)
zp_gemm_kernel(const float* __restrict__ Wpad, const float* __restrict__ RpadT,
               float* __restrict__ out) {
    __shared__ float lds[APAD * LDSW];   // 32 rows (angles) x 260 floats = 33.3 KB

    const int t   = blockIdx.x;
    const int tid = threadIdx.x;

    // Stage R^T[t] (32x256 f32, dense) into LDS rows of stride LDSW.
    const float4* src = (const float4*)(RpadT + (size_t)t * APAD * KPAD);
#pragma unroll
    for (int j0 = 0; j0 < (APAD * KPAD / 4) / 256; ++j0) {
        int j  = tid + j0 * 256;
        int a  = j >> 6;              // 64 float4 per 256-float row
        int kc = (j & 63) << 2;       // float offset within row
        *(float4*)(lds + a * LDSW + kc) = src[j];
    }
    __syncthreads();

    const int wave = tid >> 5;
    const int lane = tid & 31;
    const int half = lane >> 4;   // 0: K pair {0,1}; 1: K pair {2,3}
    const int lr   = lane & 15;

    for (int mi = 0; mi < 2; ++mi) {
        const int mt = wave * 2 + mi;                       // M tile index 0..15
        // A fragment base: row (mt*16+lr) of Wpad, K offset half*2 (8B aligned)
        const float* aptr  = Wpad + (size_t)(mt * 16 + lr) * KPAD + half * 2;
        // B fragment base in LDS^T: row n = ntile*16+lr, cols k..k+1 contiguous
        const float* b0ptr = lds + lr * LDSW + half * 2;    // N tile 0
        const float* b1ptr = b0ptr + 16 * LDSW;             // N tile 1
        __builtin_prefetch(aptr, 0, 1);

        v8f acc0 = {};
        v8f acc1 = {};
#pragma unroll 4
        for (int kt = 0; kt < KPAD / 4; ++kt) {
            v2f afr = *(const v2f*)(aptr  + kt * 4);   // global_load_b64
            v2f b0  = *(const v2f*)(b0ptr + kt * 4);   // ds_load_b64
            v2f b1  = *(const v2f*)(b1ptr + kt * 4);   // ds_load_b64
            acc0 = __builtin_amdgcn_wmma_f32_16x16x4_f32(false, afr, false, b0,
                                                         (short)0, acc0, false, false);
            acc1 = __builtin_amdgcn_wmma_f32_16x16x4_f32(false, afr, false, b1,
                                                         (short)0, acc1, false, false);
        }

        // C/D layout: VGPR r -> row mt*16 + half*8 + r, col = ntile*16 + lr
#pragma unroll
        for (int r = 0; r < 8; ++r) {
            int row = mt * 16 + half * 8 + r;
            if (row < L_LAYERS) {
                size_t base = ((size_t)t * L_LAYERS + row) * N_THETA;
                out[base + lr] = acc0[r];                       // cols 0..15
                if (16 + lr < N_THETA)                          // cols 16..29
                    out[base + 16 + lr] = acc1[r];
            }
        }
    }
}

// ---------------------------------------------------------------------------
extern "C" void kernel_launch(void* const* d_in, const int* in_sizes, int n_in,
                              void* d_out, int out_size, void* d_ws, size_t ws_size,
                              hipStream_t stream) {
    const float* vp    = (const float*)d_in[0];
    const float* vs    = (const float*)d_in[1];
    const float* rho   = (const float*)d_in[2];
    const float* theta = (const float*)d_in[3];
    const float* W     = (const float*)d_in[4];
    float*       out   = (float*)d_out;

    float* RpadT = (float*)d_ws;                                  // 600*32*256*4 = 19.66 MB
    float* Wpad  = (float*)((char*)d_ws +
                            (size_t)N_TRACES * KPAD * APAD * sizeof(float)); // +256 KB

    zp_pad_w_kernel<<<(KPAD * KPAD + 255) / 256, 256, 0, stream>>>(W, Wpad);

    const int total = N_TRACES * KPAD * APAD;
    zp_rpp_kernel<<<(total + 255) / 256, 256, 0, stream>>>(vp, vs, rho, theta, RpadT);

    zp_gemm_kernel<<<N_TRACES, 256, 0, stream>>>(Wpad, RpadT, out);
}